// NAryMarketFrameSequenceEncoder_6468220748612
// MI455X (gfx1250) — compile-verified
//
#include <hip/hip_runtime.h>
#include <math.h>

// ---------------------------------------------------------------------------
// NAC chain is linear: out = feat @ (W3*W2*W1)^T. We fuse to one 64x64 weight
// and run a bandwidth-bound streaming GEMM with V_WMMA_F32_16X16X4_F32.
// ---------------------------------------------------------------------------

typedef __attribute__((ext_vector_type(2))) float v2f;
typedef __attribute__((ext_vector_type(8))) float v8f;

#define NSYMS   256
#define NSTEPS  8192
#define NCH     62
#define NOUT    64
#define HID     16
#define DIN     64            // 2 + NCH
#define NROWS   ((long)NSYMS * NSTEPS)

#define TPB         256
#define ROWS_PER_BLK 128      // 8 waves * 16 rows
#define FSTRIDE     68        // LDS floats per feat row (64 used + 4 pad -> bank-conflict-free A frags)
#define BSTRIDE     160       // LDS floats per packed-B k-pair row (128 used + 32 pad -> half-waves on disjoint banks)

__device__ __forceinline__ float nacw(float w, float m) {
    return tanhf(w) * (1.0f / (1.0f + expf(-m)));
}

// ---------------------------------------------------------------------------
// Kernel 1: W_eff = W3 * W2 * W1 (all tanh*sigmoid transformed), written to
// workspace pre-packed in WMMA-B fragment order:
//   bpack[(k>>1)*128 + n*2 + (k&1)] = W_eff[n][k]   (B[k][n] = W_eff^T[k][n])
// so a lane's (v0,v1) = (B[k][n], B[k+1][n]) is one contiguous float2.
// ---------------------------------------------------------------------------
__global__ __launch_bounds__(TPB) void nac_fuse_weights(
    const float* __restrict__ W1h, const float* __restrict__ M1h,
    const float* __restrict__ W2h, const float* __restrict__ M2h,
    const float* __restrict__ W3h, const float* __restrict__ M3h,
    float* __restrict__ bpack)
{
    __shared__ float W1f[HID * DIN];    // 16x64
    __shared__ float W2f[HID * HID];    // 16x16
    __shared__ float W3f[NOUT * HID];   // 64x16
    __shared__ float T1 [HID * DIN];    // 16x64  = W2f @ W1f

    const int tid = threadIdx.x;
    for (int i = tid; i < HID * DIN;  i += TPB) W1f[i] = nacw(W1h[i], M1h[i]);
    for (int i = tid; i < HID * HID;  i += TPB) W2f[i] = nacw(W2h[i], M2h[i]);
    for (int i = tid; i < NOUT * HID; i += TPB) W3f[i] = nacw(W3h[i], M3h[i]);
    __syncthreads();

    // T1[h][c] = sum_k W2f[h][k] * W1f[k][c]
    for (int i = tid; i < HID * DIN; i += TPB) {
        const int h = i >> 6, c = i & 63;
        float s = 0.f;
        #pragma unroll
        for (int k = 0; k < HID; ++k) s += W2f[h * HID + k] * W1f[k * DIN + c];
        T1[i] = s;
    }
    __syncthreads();

    // W_eff[o][c] = sum_h W3f[o][h] * T1[h][c]  -> packed B layout
    for (int i = tid; i < NOUT * DIN; i += TPB) {
        const int o = i >> 6, c = i & 63;
        float s = 0.f;
        #pragma unroll
        for (int h = 0; h < HID; ++h) s += W3f[o * HID + h] * T1[h * DIN + c];
        bpack[(c >> 1) * 128 + o * 2 + (c & 1)] = s;
    }
}

// ---------------------------------------------------------------------------
// Kernel 2: out[r][n] = sym(r)*B[0][n] + t(r)*B[1][n] + sum_c mkt[r][c]*B[c+2][n]
// 8 waves/block, 16 rows per wave, K=64 via 16 x V_WMMA_F32_16X16X4_F32 per
// 16x16 output tile, 4 N-tiles. Market slab per block is contiguous ->
// coalesced staging into bank-padded LDS.
// ---------------------------------------------------------------------------
__global__ __launch_bounds__(TPB) void nac_fused_gemm(
    const float* __restrict__ market,   // [NROWS * 62], row-contiguous
    const float* __restrict__ bpack,    // [32 * 128] packed B
    float* __restrict__ out)            // [NROWS * 64]
{
    __shared__ float lf[ROWS_PER_BLK * FSTRIDE]; // 34,816 B feat tile
    __shared__ float lb[32 * BSTRIDE];           // 20,480 B packed weights

    const int  tid     = threadIdx.x;
    const long rowBase = (long)blockIdx.x * ROWS_PER_BLK;

    // Stage market slab (contiguous, coalesced) into channels [2..63]
    const float* __restrict__ mkt = market + rowBase * NCH;
    for (int i = tid; i < ROWS_PER_BLK * NCH; i += TPB) {
        const int r = i / NCH;
        const int c = i - r * NCH;
        lf[r * FSTRIDE + 2 + c] = mkt[i];
    }
    // Stage packed weights (16 KB, L2-hot) with bank padding
    for (int i = tid; i < 32 * 128; i += TPB) {
        const int kp = i >> 7, r = i & 127;
        lb[kp * BSTRIDE + r] = bpack[i];
    }
    // Synthesize (symbol, t) channels: row r -> s = r>>13, t = r&8191
    if (tid < ROWS_PER_BLK) {
        const long gr = rowBase + tid;
        lf[tid * FSTRIDE + 0] = (float)(gr >> 13);
        lf[tid * FSTRIDE + 1] = (float)(gr & (NSTEPS - 1));
    }
    __syncthreads();

    const int wave = tid >> 5;          // 0..7, one 16-row tile each
    const int lane = tid & 31;
    const int g    = lane >> 4;         // half-wave group
    const int m    = lane & 15;         // row within tile / column within N-tile

    const float* __restrict__ arow = &lf[(wave * 16 + m) * FSTRIDE];

    #pragma unroll
    for (int nt = 0; nt < 4; ++nt) {
        v8f acc = {};
        #pragma unroll
        for (int k0 = 0; k0 < DIN; k0 += 4) {
            // A 16x4 f32 frag: lanes 0-15 hold (K=k0,k0+1), lanes 16-31 (K=k0+2,k0+3)
            const v2f a = *(const v2f*)(arow + k0 + 2 * g);
            // B 4x16 f32 frag: same K split; column n = nt*16 + m
            const v2f b = *(const v2f*)(&lb[(k0 / 2 + g) * BSTRIDE + (nt * 16 + m) * 2]);
            acc = __builtin_amdgcn_wmma_f32_16x16x4_f32(
                /*neg_a=*/false, a, /*neg_b=*/false, b,
                /*c_mod=*/(short)0, acc, /*reuse_a=*/false, /*reuse_b=*/false);
        }
        // D 16x16 f32: VGPR j -> row (j + 8*g), col n = nt*16 + m
        float* __restrict__ op = out + (rowBase + wave * 16 + 8 * g) * NOUT + nt * 16 + m;
        #pragma unroll
        for (int j = 0; j < 8; ++j) op[(long)j * NOUT] = acc[j];
    }
}

extern "C" void kernel_launch(void* const* d_in, const int* in_sizes, int n_in,
                              void* d_out, int out_size, void* d_ws, size_t ws_size,
                              hipStream_t stream) {
    (void)in_sizes; (void)n_in; (void)out_size; (void)ws_size;
    const float* market = (const float*)d_in[0];
    const float* W1h    = (const float*)d_in[1];
    const float* M1h    = (const float*)d_in[2];
    const float* W2h    = (const float*)d_in[3];
    const float* M2h    = (const float*)d_in[4];
    const float* W3h    = (const float*)d_in[5];
    const float* M3h    = (const float*)d_in[6];
    float* bpack = (float*)d_ws;                 // 32*128 floats = 16 KB
    float* out   = (float*)d_out;

    nac_fuse_weights<<<1, TPB, 0, stream>>>(W1h, M1h, W2h, M2h, W3h, M3h, bpack);

    const int nblocks = (int)(NROWS / ROWS_PER_BLK);  // 16384
    nac_fused_gemm<<<nblocks, TPB, 0, stream>>>(market, bpack, out);
}